// Classifier3Stage_6064493822531
// MI455X (gfx1250) — compile-verified
//
#include <hip/hip_runtime.h>

// CDNA5 / gfx1250, wave32. Fused 3-stage routed classifier.
// One workgroup (192 threads = 6 waves) per image line h (448 lines).
// All matmuls use V_WMMA_F32_16X16X4_F32 (fp32, matches reference precision).
// Input line is staged into LDS with GLOBAL_LOAD_ASYNC_TO_LDS_B32 (ASYNCcnt).

typedef __attribute__((ext_vector_type(2))) float v2f;
typedef __attribute__((ext_vector_type(8))) float v8f;

#define HDIM 448
#define WDIM 96
#define CIND 128
#define HWDIM (HDIM * WDIM)
#define SXS 130   // sX row stride (even -> 8B-aligned pair reads, ds_load_b64)
#define WBS 34    // scratch row stride (even, padded)

__device__ __forceinline__ float lrelu(float v) { return v > 0.f ? v : 0.01f * v; }
__device__ __forceinline__ int imin(int a, int b) { return a < b ? a : b; }
__device__ __forceinline__ int imax(int a, int b) { return a > b ? a : b; }

__device__ __forceinline__ v8f wmma4(v2f a, v2f b, v8f c) {
  return __builtin_amdgcn_wmma_f32_16x16x4_f32(false, a, false, b, (short)0, c,
                                               false, false);
}

// One 16-row tile through expert cascade 128 ->(lrelu) 32 ->(lrelu) 32 -> 16
// logits, then per-row argmax over the 16 logits (lanes 0..15, one row each).
__device__ __forceinline__ int cascade_tile(
    const float* sXl, float* wbA, float* wbB, const int* sList, int ts, int cnt,
    const float* __restrict__ W0, const float* __restrict__ B0,
    const float* __restrict__ W1, const float* __restrict__ B1,
    const float* __restrict__ W2, const float* __restrict__ B2, int lane) {
  const int lr = lane & 15, lh = lane >> 4;
  const int pr = sList[ts + imin(lr, cnt - 1)];

  // layer 0: K=128, co=32; W0 is [128][32] (k-major, n contiguous)
  for (int nb = 0; nb < 32; nb += 16) {
    v8f acc = {};
    for (int k = 0; k < 128; k += 4) {
      const int ka = k + 2 * lh;
      v2f a, b;
      a.x = sXl[pr * SXS + ka];
      a.y = sXl[pr * SXS + ka + 1];
      b.x = W0[ka * 32 + nb + lr];
      b.y = W0[(ka + 1) * 32 + nb + lr];
      acc = wmma4(a, b, acc);
    }
    const float bias = B0[nb + lr];
#pragma unroll
    for (int v = 0; v < 8; ++v)
      wbA[(v + 8 * lh) * WBS + nb + lr] = lrelu(acc[v] + bias);
  }
  asm volatile("s_wait_dscnt 0" ::: "memory");

  // layer 1: K=32, co=32; W1 is [32][32]
  for (int nb = 0; nb < 32; nb += 16) {
    v8f acc = {};
    for (int k = 0; k < 32; k += 4) {
      const int ka = k + 2 * lh;
      v2f a, b;
      a.x = wbA[lr * WBS + ka];
      a.y = wbA[lr * WBS + ka + 1];
      b.x = W1[ka * 32 + nb + lr];
      b.y = W1[(ka + 1) * 32 + nb + lr];
      acc = wmma4(a, b, acc);
    }
    const float bias = B1[nb + lr];
#pragma unroll
    for (int v = 0; v < 8; ++v)
      wbB[(v + 8 * lh) * WBS + nb + lr] = lrelu(acc[v] + bias);
  }
  asm volatile("s_wait_dscnt 0" ::: "memory");

  // layer 2: K=32, co=16 logits; W2 is [32][16]
  {
    v8f acc = {};
    for (int k = 0; k < 32; k += 4) {
      const int ka = k + 2 * lh;
      v2f a, b;
      a.x = wbB[lr * WBS + ka];
      a.y = wbB[lr * WBS + ka + 1];
      b.x = W2[ka * 16 + lr];
      b.y = W2[(ka + 1) * 16 + lr];
      acc = wmma4(a, b, acc);
    }
    const float bias = B2[lr];
#pragma unroll
    for (int v = 0; v < 8; ++v)
      wbA[(v + 8 * lh) * WBS + lr] = acc[v] + bias;
  }
  asm volatile("s_wait_dscnt 0" ::: "memory");

  // per-row argmax over 16 logits (first-max wins, matches jnp.argmax)
  int best = 0;
  if (lane < 16) {
    float bv = wbA[lane * WBS + 0];
    for (int c = 1; c < 16; ++c) {
      float v = wbA[lane * WBS + c];
      if (v > bv) { bv = v; best = c; }
    }
  }
  return best;
}

__global__ __launch_bounds__(192, 1) void c3s_kernel(
    const float* __restrict__ x_in,
    const float* __restrict__ w1_0, const float* __restrict__ b1_0,
    const float* __restrict__ w1_1, const float* __restrict__ b1_1,
    const float* __restrict__ w1_2, const float* __restrict__ b1_2,
    const float* __restrict__ w2_0, const float* __restrict__ b2_0,
    const float* __restrict__ w2_1, const float* __restrict__ b2_1,
    const float* __restrict__ w2_2, const float* __restrict__ b2_2,
    const float* __restrict__ w3_0, const float* __restrict__ b3_0,
    const float* __restrict__ w3_1, const float* __restrict__ b3_1,
    const float* __restrict__ w3_2, const float* __restrict__ b3_2,
    int* __restrict__ out) {
  __shared__ float sX[96 * SXS];       // line activations, pixel-major, padded
  __shared__ float sA[96 * WBS];       // stage-1 ping
  __shared__ float sB[96 * WBS];       // stage-1 pong
  __shared__ float sWB[6][16 * WBS];   // per-wave tile scratch a
  __shared__ float sWC[6][16 * WBS];   // per-wave tile scratch b
  __shared__ int sInds1[96], sI12r[96], sI12c[96];
  __shared__ int sCnt[64], sPos[64], sList[96];
  __shared__ int sTileE[96], sTileS[96], sTileC[96];
  __shared__ int sNT;

  const int h = blockIdx.x;
  const int tid = threadIdx.x;
  const int lane = tid & 31;
  const int wv = tid >> 5;
  const int lr = lane & 15, lh = lane >> 4;

  // Stage input: x_in is [C][H][W]; transpose the line into [w][c] in LDS
  // using async global->LDS copies (no VGPR round-trip, ASYNCcnt-tracked).
  for (int i = tid; i < 96 * 128; i += 192) {
    const int c = i / 96, w = i - c * 96;
    const float* gaddr = x_in + (size_t)c * HWDIM + (size_t)h * 96 + w;
    const unsigned laddr = (unsigned)(uintptr_t)(&sX[w * SXS + c]);
    asm volatile("global_load_async_to_lds_b32 %0, %1, off"
                 :: "v"(laddr), "v"(gaddr) : "memory");
  }
  asm volatile("s_wait_asynccnt 0" ::: "memory");
  __syncthreads();

  // ---------------- stage 1: per-line grouped conv (dense WMMA GEMMs) ------
  {  // layer 0: 128 -> 32 ; w1_0 is (H,32,128) i.e. [n][k]
    const int mb = wv * 16;
    const float* W0 = w1_0 + (size_t)h * 32 * 128;
    const float* B0 = b1_0 + h * 32;
    for (int nb = 0; nb < 32; nb += 16) {
      v8f acc = {};
      for (int k = 0; k < 128; k += 4) {
        const int ka = k + 2 * lh;
        v2f a, b;
        a.x = sX[(mb + lr) * SXS + ka];
        a.y = sX[(mb + lr) * SXS + ka + 1];
        b.x = W0[(nb + lr) * 128 + ka];
        b.y = W0[(nb + lr) * 128 + ka + 1];
        acc = wmma4(a, b, acc);
      }
      const float bias = B0[nb + lr];
#pragma unroll
      for (int v = 0; v < 8; ++v)
        sA[(mb + v + 8 * lh) * WBS + nb + lr] = lrelu(acc[v] + bias);
    }
  }
  __syncthreads();
  {  // layer 1: 32 -> 32
    const int mb = wv * 16;
    const float* W1 = w1_1 + (size_t)h * 32 * 32;
    const float* B1 = b1_1 + h * 32;
    for (int nb = 0; nb < 32; nb += 16) {
      v8f acc = {};
      for (int k = 0; k < 32; k += 4) {
        const int ka = k + 2 * lh;
        v2f a, b;
        a.x = sA[(mb + lr) * WBS + ka];
        a.y = sA[(mb + lr) * WBS + ka + 1];
        b.x = W1[(nb + lr) * 32 + ka];
        b.y = W1[(nb + lr) * 32 + ka + 1];
        acc = wmma4(a, b, acc);
      }
      const float bias = B1[nb + lr];
#pragma unroll
      for (int v = 0; v < 8; ++v)
        sB[(mb + v + 8 * lh) * WBS + nb + lr] = lrelu(acc[v] + bias);
    }
  }
  __syncthreads();
  {  // layer 2: 32 -> 8 logits (single n-tile, upper 8 columns zeroed)
    const int mb = wv * 16;
    const float* W2 = w1_2 + (size_t)h * 8 * 32;
    const float* B2 = b1_2 + h * 8;
    v8f acc = {};
    for (int k = 0; k < 32; k += 4) {
      const int ka = k + 2 * lh;
      v2f a, b;
      a.x = sB[(mb + lr) * WBS + ka];
      a.y = sB[(mb + lr) * WBS + ka + 1];
      b.x = (lr < 8) ? W2[lr * 32 + ka] : 0.f;
      b.y = (lr < 8) ? W2[lr * 32 + ka + 1] : 0.f;
      acc = wmma4(a, b, acc);
    }
    if (lr < 8) {
      const float bias = B2[lr];
#pragma unroll
      for (int v = 0; v < 8; ++v)
        sA[(mb + v + 8 * lh) * WBS + lr] = acc[v] + bias;
    }
  }
  __syncthreads();
  if (tid < 96) {  // argmax over 8 -> inds1
    float bv = sA[tid * WBS];
    int bi = 0;
    for (int c = 1; c < 8; ++c) {
      float v = sA[tid * WBS + c];
      if (v > bv) { bv = v; bi = c; }
    }
    sInds1[tid] = bi;
  }
  __syncthreads();

  // ---------------- stage 2: bucket by inds1 (8 experts/line), tiled WMMA --
  if (tid < 64) sCnt[tid] = 0;
  __syncthreads();
  if (tid < 96) atomicAdd(&sCnt[sInds1[tid]], 1);
  __syncthreads();
  if (tid == 0) {
    int off = 0, nt = 0;
    for (int e = 0; e < 8; ++e) {
      sPos[e] = off;
      const int c = sCnt[e];
      for (int s = 0; s < c; s += 16) {
        sTileE[nt] = e; sTileS[nt] = off + s; sTileC[nt] = imin(16, c - s); ++nt;
      }
      off += c;
    }
    sNT = nt;
  }
  __syncthreads();
  if (tid < 96) { const int p = atomicAdd(&sPos[sInds1[tid]], 1); sList[p] = tid; }
  __syncthreads();

  for (int t = wv; t < sNT; t += 6) {
    const int e = sTileE[t], ts = sTileS[t], cnt = sTileC[t];
    const long eg = (long)h * 8 + e;
    const int best = cascade_tile(sX, sWB[wv], sWC[wv], sList, ts, cnt,
                                  w2_0 + eg * (128 * 32), b2_0 + eg * 32,
                                  w2_1 + eg * (32 * 32), b2_1 + eg * 32,
                                  w2_2 + eg * (32 * 16), b2_2 + eg * 16, lane);
    if (lane < 16 && lane < cnt) {
      const int p = sList[ts + lane];
      const int raw = sInds1[p] * 8 + (best - 4);   // inds12 (unclipped)
      sI12r[p] = raw;
      sI12c[p] = imin(imax(raw, 0), 63);            // clipped for routing
    }
  }
  __syncthreads();

  // ---------------- stage 3: bucket by clipped inds12 (64 experts/line) ----
  if (tid < 64) sCnt[tid] = 0;
  __syncthreads();
  if (tid < 96) atomicAdd(&sCnt[sI12c[tid]], 1);
  __syncthreads();
  if (tid == 0) {
    int off = 0, nt = 0;
    for (int e = 0; e < 64; ++e) {
      sPos[e] = off;
      const int c = sCnt[e];
      for (int s = 0; s < c; s += 16) {
        sTileE[nt] = e; sTileS[nt] = off + s; sTileC[nt] = imin(16, c - s); ++nt;
      }
      off += c;
    }
    sNT = nt;
  }
  __syncthreads();
  if (tid < 96) { const int p = atomicAdd(&sPos[sI12c[tid]], 1); sList[p] = tid; }
  __syncthreads();

  for (int t = wv; t < sNT; t += 6) {
    const int e = sTileE[t], ts = sTileS[t], cnt = sTileC[t];
    const long eg = (long)h * 64 + e;  // DIV==1: inds12_s == inds12_l
    const int best = cascade_tile(sX, sWB[wv], sWC[wv], sList, ts, cnt,
                                  w3_0 + eg * (128 * 32), b3_0 + eg * 32,
                                  w3_1 + eg * (32 * 32), b3_1 + eg * 32,
                                  w3_2 + eg * (32 * 16), b3_2 + eg * 16, lane);
    if (lane < 16 && lane < cnt) {
      const int p = sList[ts + lane];
      const int fin = sI12r[p] * 8 + (best - 4);    // uses UNclipped inds12
      out[h * 96 + p] = imin(imax(fin, 0), 511);
    }
  }
}

extern "C" void kernel_launch(void* const* d_in, const int* in_sizes, int n_in,
                              void* d_out, int out_size, void* d_ws,
                              size_t ws_size, hipStream_t stream) {
  const float* x_in = (const float*)d_in[0];
  const float* w1_0 = (const float*)d_in[1];
  const float* b1_0 = (const float*)d_in[2];
  const float* w1_1 = (const float*)d_in[3];
  const float* b1_1 = (const float*)d_in[4];
  const float* w1_2 = (const float*)d_in[5];
  const float* b1_2 = (const float*)d_in[6];
  const float* w2_0 = (const float*)d_in[7];
  const float* b2_0 = (const float*)d_in[8];
  const float* w2_1 = (const float*)d_in[9];
  const float* b2_1 = (const float*)d_in[10];
  const float* w2_2 = (const float*)d_in[11];
  const float* b2_2 = (const float*)d_in[12];
  const float* w3_0 = (const float*)d_in[13];
  const float* b3_0 = (const float*)d_in[14];
  const float* w3_1 = (const float*)d_in[15];
  const float* b3_1 = (const float*)d_in[16];
  const float* w3_2 = (const float*)d_in[17];
  const float* b3_2 = (const float*)d_in[18];
  int* out = (int*)d_out;

  c3s_kernel<<<dim3(HDIM), dim3(192), 0, stream>>>(
      x_in, w1_0, b1_0, w1_1, b1_1, w1_2, b1_2, w2_0, b2_0, w2_1, b2_1, w2_2,
      b2_2, w3_0, b3_0, w3_1, b3_1, w3_2, b3_2, out);
}